// ProposedModel_1967095022103
// MI455X (gfx1250) — compile-verified
//
#include <hip/hip_runtime.h>
#include <math.h>

#define NN 8192

typedef float v2f __attribute__((ext_vector_type(2)));
typedef float v8f __attribute__((ext_vector_type(8)));

// ---------------------------------------------------------------------------
// Kernel 1: h = W @ x + b   (fp32 GEMV, HBM-bandwidth bound: 256MB / 23.3TB/s)
// Uses V_WMMA_F32_16X16X4_F32. One wave owns a 16-row tile of W; the 4 waves
// of a block split the K dimension and reduce through LDS.
//
// A (16x4 f32) per-lane layout: lane L -> row (L&15); VGPR0=K(2*(L>>4)),
// VGPR1=K(2*(L>>4)+1)  => a single float2 load per lane per WMMA.
// B (4x16)   : row k striped across lanes in one VGPR; we broadcast x[k] to
// all 16 columns so every column of D equals the y-tile.
// ---------------------------------------------------------------------------
__global__ __launch_bounds__(128) void gemv_wmma_f32(
    const float* __restrict__ W, const float* __restrict__ x,
    const float* __restrict__ b, float* __restrict__ h)
{
  __shared__ float partial[4][16];
  const int tid  = threadIdx.x;
  const int wave = tid >> 5;
  const int lane = tid & 31;
  const int half = lane >> 4;   // which K-pair this lane holds
  const int mrow = lane & 15;   // row within the 16-row tile
  const long row = (long)blockIdx.x * 16 + mrow;
  const float* __restrict__ wrow = W + row * (long)NN;

  v8f acc = {};
  const int kBeg = wave * (NN / 4);
  const int kEnd = kBeg + (NN / 4);
#pragma unroll 4
  for (int k = kBeg; k < kEnd; k += 4) {
    if ((k & 255) == 0) __builtin_prefetch(wrow + k + 1024, 0, 1);
    v2f a  = *(const v2f*)(wrow + k + 2 * half);  // A: W[row][k+2h .. k+2h+1]
    v2f bv = *(const v2f*)(x    + k + 2 * half);  // B: x broadcast to all N
    acc = __builtin_amdgcn_wmma_f32_16x16x4_f32(
        false, a, false, bv, (short)0, acc, false, false);
  }

  // D layout: VGPR r, lanes 0-15 -> M=r; lanes 16-31 -> M=r+8 (all N equal).
  if (lane == 0) {
#pragma unroll
    for (int r2 = 0; r2 < 8; ++r2) partial[wave][r2] = acc[r2];
  } else if (lane == 16) {
#pragma unroll
    for (int r2 = 0; r2 < 8; ++r2) partial[wave][8 + r2] = acc[r2];
  }
  __syncthreads();
  if (tid < 16) {
    int o = blockIdx.x * 16 + tid;
    h[o] = partial[0][tid] + partial[1][tid] + partial[2][tid] +
           partial[3][tid] + b[o];
  }
}

// ---------------------------------------------------------------------------
// Kernel 2: budget-constrained softmax over 8192 elements. Single workgroup
// (1024 threads, one WGP): LDS bitonic sort of (key, idx), then register-
// blocked scans (prefix sum of cs, suffix logsumexp of xs), block reductions,
// and scatter back through the sorted index. Fully deterministic.
// LDS: 32KB keys + 16KB u16 idx + 4KB scan aux.
// ---------------------------------------------------------------------------
#define T2 1024
#define PERT 8

__device__ __forceinline__ float lse2(float a, float bb) {
  if (bb == -INFINITY) return a;
  if (a  == -INFINITY) return bb;
  float m = fmaxf(a, bb);
  return m + logf(expf(a - m) + expf(bb - m));
}

__global__ __launch_bounds__(1024) void bcsoftmax1d_kernel(
    const float* __restrict__ h, const float* __restrict__ c,
    float* __restrict__ out)
{
  __shared__ float          skey[NN];
  __shared__ unsigned short sidx[NN];
  __shared__ float          aux[T2];

  const int tid = threadIdx.x;

  // keys: log(c) - h, ascending sort, stable via index tie-break
  for (int i = tid; i < NN; i += T2) {
    skey[i] = logf(c[i]) - h[i];
    sidx[i] = (unsigned short)i;
  }
  __syncthreads();

  // bitonic sort (91 stages for 2^13)
  for (int k = 2; k <= NN; k <<= 1) {
    for (int j = k >> 1; j > 0; j >>= 1) {
      for (int t = tid; t < NN; t += T2) {
        int p = t ^ j;
        if (p > t) {
          float ka = skey[t], kb = skey[p];
          unsigned short ia = sidx[t], ib = sidx[p];
          bool up   = ((t & k) == 0);
          bool agtb = (ka > kb) || ((ka == kb) && (ia > ib));
          if (agtb == up) {
            skey[t] = kb; skey[p] = ka;
            sidx[t] = ib; sidx[p] = ia;
          }
        }
      }
      __syncthreads();
    }
  }

  // each thread owns 8 consecutive sorted positions
  const int base = tid * PERT;
  float xs8[PERT], cs8[PERT], s8[PERT], lr8[PERT];
  bool  kb8[PERT];
#pragma unroll
  for (int p = 0; p < PERT; ++p) {
    int gi = sidx[base + p];
    xs8[p] = h[gi];
    cs8[p] = c[gi];
  }

  // s[i] = 1 - exclusive_prefix_sum(cs)[i]
  float run = 0.f;
#pragma unroll
  for (int p = 0; p < PERT; ++p) { s8[p] = run; run += cs8[p]; }
  aux[tid] = run;
  __syncthreads();
  for (int off = 1; off < T2; off <<= 1) {       // Hillis-Steele inclusive
    float v = (tid >= off) ? aux[tid - off] : 0.f;
    __syncthreads();
    aux[tid] += v;
    __syncthreads();
  }
  float thrExcl = aux[tid] - run;
  __syncthreads();
#pragma unroll
  for (int p = 0; p < PERT; ++p) s8[p] = 1.0f - (thrExcl + s8[p]);

  // log_r[i] = logsumexp(xs[i:])  (inclusive suffix scan with LSE combine)
  float suf = -INFINITY;
#pragma unroll
  for (int p = PERT - 1; p >= 0; --p) { suf = lse2(xs8[p], suf); lr8[p] = suf; }
  aux[tid] = suf;
  __syncthreads();
  for (int off = 1; off < T2; off <<= 1) {
    float v = (tid + off < T2) ? aux[tid + off] : -INFINITY;
    __syncthreads();
    aux[tid] = lse2(aux[tid], v);
    __syncthreads();
  }
  float after = (tid + 1 < T2) ? aux[tid + 1] : -INFINITY;
  __syncthreads();
#pragma unroll
  for (int p = 0; p < PERT; ++p) lr8[p] = lse2(lr8[p], after);

  // in_kb, then block reductions: m (masked max), sum(cs*in_kb), r
  float lmax = -INFINITY, lkb = 0.f;
#pragma unroll
  for (int p = 0; p < PERT; ++p) {
    bool kb = (cs8[p] == 0.f) ||
              (((s8[p] - cs8[p]) > 0.f) &&
               ((xs8[p] - lr8[p] + logf(s8[p])) > logf(cs8[p])));
    kb8[p] = kb;
    if (!kb) lmax = fmaxf(lmax, xs8[p]);
    else     lkb += cs8[p];
  }
  aux[tid] = lmax;
  __syncthreads();
  for (int off = T2 / 2; off > 0; off >>= 1) {
    if (tid < off) aux[tid] = fmaxf(aux[tid], aux[tid + off]);
    __syncthreads();
  }
  float m = aux[0];
  __syncthreads();
  aux[tid] = lkb;
  __syncthreads();
  for (int off = T2 / 2; off > 0; off >>= 1) {
    if (tid < off) aux[tid] += aux[tid + off];
    __syncthreads();
  }
  float s2 = 1.0f - aux[0];
  __syncthreads();
  float lr = 0.f;
#pragma unroll
  for (int p = 0; p < PERT; ++p) if (!kb8[p]) lr += expf(xs8[p] - m);
  aux[tid] = lr;
  __syncthreads();
  for (int off = T2 / 2; off > 0; off >>= 1) {
    if (tid < off) aux[tid] += aux[tid + off];
    __syncthreads();
  }
  float r = aux[0];

  // ys and scatter back through the sort permutation
#pragma unroll
  for (int p = 0; p < PERT; ++p) {
    int gi  = sidx[base + p];
    float y = kb8[p] ? cs8[p] : (s2 * expf(xs8[p] - m) / r);
    out[gi] = y;
  }
}

// ---------------------------------------------------------------------------
extern "C" void kernel_launch(void* const* d_in, const int* in_sizes, int n_in,
                              void* d_out, int out_size, void* d_ws, size_t ws_size,
                              hipStream_t stream) {
  const float* x = (const float*)d_in[0];   // (8192,)
  const float* c = (const float*)d_in[1];   // (8192,)
  const float* W = (const float*)d_in[2];   // (8192, 8192)
  const float* b = (const float*)d_in[3];   // (8192,)
  float* out = (float*)d_out;
  float* h   = (float*)d_ws;                // 8192 floats of scratch

  gemv_wmma_f32<<<NN / 16, 128, 0, stream>>>(W, x, b, h);
  bcsoftmax1d_kernel<<<1, T2, 0, stream>>>(h, c, out);
}